// Wkv7Op_26740466385131
// MI455X (gfx1250) — compile-verified
//
#include <hip/hip_runtime.h>

#define TSEQ 2048
#define NH 32
#define NN 64
#define CC 16
#define NCHUNK (TSEQ / CC)

typedef __attribute__((ext_vector_type(2))) float v2f;
typedef __attribute__((ext_vector_type(8))) float v8f;
typedef __attribute__((ext_vector_type(4))) unsigned int v4u;
typedef __attribute__((ext_vector_type(8))) int v8i;
typedef __attribute__((ext_vector_type(4))) int v4i;

__device__ __forceinline__ v8f wmma4(v2f a, v2f b, v8f c) {
  // D = A(16x4) x B(4x16) + C(16x16), fp32, wave32
  return __builtin_amdgcn_wmma_f32_16x16x4_f32(false, a, false, b, (short)0, c,
                                               false, false);
}

__device__ __forceinline__ v8f zero8() {
  v8f z;
#pragma unroll
  for (int p = 0; p < 8; ++p) z[p] = 0.0f;
  return z;
}

// ---- WMMA fragment loaders (gfx1250 f32 layouts) ----
__device__ __forceinline__ v2f load_a_rm(const float* M, int ld, int m0, int k0, int lane) {
  int m = lane & 15, kh = (lane >> 4) << 1;
  v2f x;
  x.x = M[(m0 + m) * ld + k0 + kh];
  x.y = M[(m0 + m) * ld + k0 + kh + 1];
  return x;
}
__device__ __forceinline__ v2f load_a_tr(const float* M, int ld, int m0, int k0, int lane) {
  int m = lane & 15, kh = (lane >> 4) << 1;
  v2f x;
  x.x = M[(k0 + kh) * ld + m0 + m];
  x.y = M[(k0 + kh + 1) * ld + m0 + m];
  return x;
}
__device__ __forceinline__ v2f load_b_rm(const float* M, int ld, int k0, int n0, int lane) {
  int n = lane & 15, kh = (lane >> 4) << 1;
  v2f x;
  x.x = M[(k0 + kh) * ld + n0 + n];
  x.y = M[(k0 + kh + 1) * ld + n0 + n];
  return x;
}
__device__ __forceinline__ v2f load_b_tr(const float* M, int ld, int k0, int n0, int lane) {
  int n = lane & 15, kh = (lane >> 4) << 1;
  v2f x;
  x.x = M[(n0 + n) * ld + k0 + kh];
  x.y = M[(n0 + n) * ld + k0 + kh + 1];
  return x;
}
__device__ __forceinline__ v8f load_c(const float* M, int ld, int m0, int n0, int lane) {
  int n = lane & 15, mh = (lane >> 4) << 3;
  v8f x;
#pragma unroll
  for (int p = 0; p < 8; ++p) x[p] = M[(m0 + mh + p) * ld + n0 + n];
  return x;
}
__device__ __forceinline__ void store_d(float* M, int ld, int m0, int n0, int lane, v8f d) {
  int n = lane & 15, mh = (lane >> 4) << 3;
#pragma unroll
  for (int p = 0; p < 8; ++p) M[(m0 + mh + p) * ld + n0 + n] = d[p];
}

// ---- TDM: async 2D tile load (16 rows x 64 f32, row stride NH*NN) into LDS ----
__device__ __forceinline__ void tdm_load_2d(const float* gptr, float* lptr) {
  unsigned long long ga = (unsigned long long)gptr;
  unsigned int lds = (unsigned int)(unsigned long long)(void*)lptr;  // low 32b = LDS byte offset
  v4u g0;
  g0[0] = 1u;                                        // count=1, user D#
  g0[1] = lds;                                       // lds_addr
  g0[2] = (unsigned int)(ga & 0xFFFFFFFFu);          // global_addr[31:0]
  g0[3] = (unsigned int)((ga >> 32) & 0x01FFFFFFu)   // global_addr[56:32]
          | (2u << 30);                              // type = 2 ("image")
  v8i g1;
  g1[0] = 0x00020000;            // data_size=2 (4B); mask=0; no pad/iter/barrier
  g1[1] = (int)(64u << 16);      // tensor_dim0 = 64   (bits [79:48] lo)
  g1[2] = (int)(16u << 16);      // tensor_dim0 hi = 0; tensor_dim1 = 16 (lo)
  g1[3] = (int)(64u << 16);      // tensor_dim1 hi = 0; tile_dim0 = 64
  g1[4] = 16;                    // tile_dim1 = 16; tile_dim2 = 0
  g1[5] = (int)(NH * NN);        // tensor_dim0_stride = 2048 (data_size units)
  g1[6] = 0;                     // stride hi; tensor_dim1_stride lo
  g1[7] = 0;
  v4i g2 = {0, 0, 0, 0};
  v4i g3 = {0, 0, 0, 0};
#if defined(__clang_major__) && (__clang_major__ >= 23)
  v8i g4 = {0, 0, 0, 0, 0, 0, 0, 0};
  __builtin_amdgcn_tensor_load_to_lds(g0, g1, g2, g3, g4, 0);
#else
  __builtin_amdgcn_tensor_load_to_lds(g0, g1, g2, g3, 0);
#endif
}

__global__ __launch_bounds__(128) void wkv7_chunked_kernel(
    const float* __restrict__ r, const float* __restrict__ w,
    const float* __restrict__ k, const float* __restrict__ v,
    const float* __restrict__ a, const float* __restrict__ b,
    const float* __restrict__ s0, float* __restrict__ y_out,
    float* __restrict__ s_out) {
  __shared__ float S[NN * NN];                           // state [v][key]
  __shared__ float rc[2][CC * NN], wc_[2][CC * NN], kc[2][CC * NN];
  __shared__ float vc[2][CC * NN], ac[2][CC * NN], bc[2][CC * NN];
  __shared__ float At[CC * NN];                          // a~_s = cumW(<s) * a_s
  __shared__ float Rt[CC * NN];                          // cumW(<=t) * r_t
  __shared__ float Kh[CC * NN];                          // P_{s+1,t}^T k_s
  __shared__ float Bh[CC * NN];                          // P_{s+1,t}^T b_s
  __shared__ float Wf[NN];                               // full-chunk cumprod of w
  __shared__ float Gm[CC * CC], Hm[CC * CC];             // triangular scalars
  __shared__ float U[NN * CC];                           // S0 @ Atilde
  __shared__ float dka[CC], dba[CC];

  const int h = blockIdx.x;
  const int tid = threadIdx.x;
  const int lane = tid & 31;
  const int wv = tid >> 5;  // wave id -> M-block (rows 16*wv .. 16*wv+15)

  for (int i = tid; i < NN * NN; i += 128) S[i] = s0[h * NN * NN + i];

  // ---- prefetch chunk 0 via TDM (wave 0 only; TDM issues once per wave) ----
  if (wv == 0) {
    const int g0off = (0 * NH + h) * NN;
    tdm_load_2d(r + g0off, rc[0]);
    tdm_load_2d(w + g0off, wc_[0]);
    tdm_load_2d(k + g0off, kc[0]);
    tdm_load_2d(v + g0off, vc[0]);
    tdm_load_2d(a + g0off, ac[0]);
    tdm_load_2d(b + g0off, bc[0]);
  }
  __syncthreads();

  for (int c = 0; c < NCHUNK; ++c) {
    const int buf = c & 1;
    const float* rcb = rc[buf];
    const float* wcb = wc_[buf];
    const float* kcb = kc[buf];
    const float* vcb = vc[buf];
    const float* acb = ac[buf];
    const float* bcb = bc[buf];

    // ---- wait for this chunk's TDM tiles, then prefetch next chunk ----
    if (wv == 0) __builtin_amdgcn_s_wait_tensorcnt(0);
    __syncthreads();
    if (wv == 0 && (c + 1) < NCHUNK) {
      const int nb = (c + 1) & 1;
      const int goff = (((c + 1) * CC) * NH + h) * NN;
      tdm_load_2d(r + goff, rc[nb]);
      tdm_load_2d(w + goff, wc_[nb]);
      tdm_load_2d(k + goff, kc[nb]);
      tdm_load_2d(v + goff, vc[nb]);
      tdm_load_2d(a + goff, ac[nb]);
      tdm_load_2d(b + goff, bc[nb]);
    }

    // ---- phase 2: per-channel cumprods (64 lanes) + zero G/H (rest) ----
    if (tid < NN) {
      float cp = 1.0f;
      for (int t = 0; t < CC; ++t) {
        At[t * NN + tid] = cp * acb[t * NN + tid];
        cp *= wcb[t * NN + tid];
        Rt[t * NN + tid] = cp * rcb[t * NN + tid];
      }
      Wf[tid] = cp;
    } else {
      for (int i = tid - NN; i < CC * CC; i += 64) { Gm[i] = 0.0f; Hm[i] = 0.0f; }
    }
    __syncthreads();

    // ---- phase 3: O(C^2 N) sequential within-chunk recurrences ----
    for (int t = 0; t < CC; ++t) {
      if (tid < CC) {
        int s = tid;
        if (s < t) {
          float d = 0.0f;
          for (int n = 0; n < NN; ++n) d += acb[t * NN + n] * Kh[s * NN + n];
          dka[s] = d;
        }
      } else if (tid < 2 * CC) {
        int s = tid - CC;
        if (s < t) {
          float d = 0.0f;
          for (int n = 0; n < NN; ++n) d += acb[t * NN + n] * Bh[s * NN + n];
          dba[s] = d;
        }
      }
      __syncthreads();
      for (int i = tid; i < t * NN; i += 128) {
        int s = i >> 6, n = i & 63;
        Kh[i] = wcb[t * NN + n] * Kh[i] + bcb[t * NN + n] * dka[s];
        Bh[i] = wcb[t * NN + n] * Bh[i] + bcb[t * NN + n] * dba[s];
      }
      if (tid < NN) {
        Kh[t * NN + tid] = kcb[t * NN + tid];
        Bh[t * NN + tid] = bcb[t * NN + tid];
      }
      __syncthreads();
      if (tid < CC) {
        int s = tid;
        if (s <= t) {
          float d = 0.0f;
          for (int n = 0; n < NN; ++n) d += Kh[s * NN + n] * rcb[t * NN + n];
          Gm[s * CC + t] = d;
        }
      } else if (tid < 2 * CC) {
        int s = tid - CC;
        if (s <= t) {
          float d = 0.0f;
          for (int n = 0; n < NN; ++n) d += Bh[s * NN + n] * rcb[t * NN + n];
          Hm[s * CC + t] = d;
        }
      }
      __syncthreads();
    }

    // ---- phase 4 (WMMA): U = S0 @ Atilde ; Ypart = S0 @ Rtilde ----
    v8f cu = zero8(), cy = zero8();
#pragma unroll
    for (int kk = 0; kk < NN; kk += 4) {
      v2f aS = load_a_rm(S, NN, 16 * wv, kk, lane);
      cu = wmma4(aS, load_b_tr(At, NN, kk, 0, lane), cu);
      cy = wmma4(aS, load_b_tr(Rt, NN, kk, 0, lane), cy);
    }
    store_d(U, CC, 16 * wv, 0, lane, cu);
    __syncthreads();

    // ---- phase 5 (WMMA): Y += U*H + V*G ; store y straight to global ----
#pragma unroll
    for (int kk = 0; kk < CC; kk += 4) {
      cy = wmma4(load_a_rm(U, CC, 16 * wv, kk, lane),
                 load_b_rm(Hm, CC, kk, 0, lane), cy);
      cy = wmma4(load_a_tr(vcb, NN, 16 * wv, kk, lane),
                 load_b_rm(Gm, CC, kk, 0, lane), cy);
    }
    {
      int n = lane & 15, mh = (lane >> 4) << 3;  // D tile: m = channel, n = t
#pragma unroll
      for (int p = 0; p < 8; ++p)
        y_out[((c * CC + n) * NH + h) * NN + 16 * wv + mh + p] = cy[p];
    }

    // ---- phase 6 (WMMA): S = S (x) colW + U*Bh^T + V*Kh^T (own M-block) ----
    for (int e = lane; e < 16 * NN; e += 32) {
      int rr = e >> 6, n = e & 63;
      S[(16 * wv + rr) * NN + n] *= Wf[n];
    }
#pragma unroll
    for (int nb2 = 0; nb2 < 4; ++nb2) {
      v8f cs = load_c(S, NN, 16 * wv, 16 * nb2, lane);
#pragma unroll
      for (int kk = 0; kk < CC; kk += 4) {
        cs = wmma4(load_a_rm(U, CC, 16 * wv, kk, lane),
                   load_b_rm(Bh, NN, kk, 16 * nb2, lane), cs);
        cs = wmma4(load_a_tr(vcb, NN, 16 * wv, kk, lane),
                   load_b_rm(Kh, NN, kk, 16 * nb2, lane), cs);
      }
      store_d(S, NN, 16 * wv, 16 * nb2, lane, cs);
    }
    __syncthreads();
  }

  for (int i = tid; i < NN * NN; i += 128) s_out[h * NN * NN + i] = S[i];
}

extern "C" void kernel_launch(void* const* d_in, const int* in_sizes, int n_in,
                              void* d_out, int out_size, void* d_ws, size_t ws_size,
                              hipStream_t stream) {
  const float* r = (const float*)d_in[0];
  const float* w = (const float*)d_in[1];
  const float* k = (const float*)d_in[2];
  const float* v = (const float*)d_in[3];
  const float* a = (const float*)d_in[4];
  const float* b = (const float*)d_in[5];
  const float* s0 = (const float*)d_in[6];
  float* y = (float*)d_out;
  float* sf = (float*)d_out + (size_t)TSEQ * NH * NN;
  wkv7_chunked_kernel<<<dim3(NH), dim3(128), 0, stream>>>(r, w, k, v, a, b, s0, y, sf);
}